// CombinedRotaryEmbedding_89318139887730
// MI455X (gfx1250) — compile-verified
//
#include <hip/hip_runtime.h>
#include <hip/hip_bf16.h>

// Problem constants (match reference)
#define B_DIM 8
#define S_DIM 4096
#define N_STATE 1024
#define N_HEAD 16
#define H_DIM 64
#define NUM_ROT 32

typedef __attribute__((ext_vector_type(2))) float v2f;
typedef __attribute__((ext_vector_type(8))) float v8f;

// ---------------------------------------------------------------------------
// Kernel 1: build M = (product of Givens rotations) @ rotation_matrix  (64x64)
// One block of 64 threads; thread t owns row t of R.
// ---------------------------------------------------------------------------
__global__ __launch_bounds__(64) void build_M_kernel(
    const float* __restrict__ thetas,
    const float* __restrict__ pairs,       // (NUM_ROT, 2) float
    const float* __restrict__ theta_scale, // (1,)
    const float* __restrict__ rotmat,      // (64,64)
    float* __restrict__ M_out)             // (64,64)
{
    __shared__ float R[H_DIM][H_DIM];
    const int t = threadIdx.x; // row 0..63
    #pragma unroll 4
    for (int c = 0; c < H_DIM; ++c) R[t][c] = (t == c) ? 1.0f : 0.0f;
    const float ts = theta_scale[0];
    for (int step = 0; step < NUM_ROT; ++step) {
        const int i = (int)pairs[2 * step + 0];
        const int j = (int)pairs[2 * step + 1];
        const float th = thetas[step] * ts;
        const float c = __cosf(th);
        const float s = __sinf(th);
        const float ri = R[t][i];
        const float rj = R[t][j];
        if (i != j) {
            R[t][i] = c * ri + s * rj;   // new col i
            R[t][j] = -s * ri + c * rj;  // new col j
        } else {
            // reference overwrite semantics: G collapses to diag with (i,i)=c
            R[t][i] = c * ri;
        }
    }
    __syncthreads();
    // M = R @ rotmat ; thread t computes row t
    #pragma unroll 1
    for (int c = 0; c < H_DIM; ++c) {
        float acc = 0.0f;
        #pragma unroll 8
        for (int k = 0; k < H_DIM; ++k) acc = fmaf(R[t][k], rotmat[k * H_DIM + c], acc);
        M_out[t * H_DIM + c] = acc;
    }
}

// ---------------------------------------------------------------------------
// Kernel 2: fused  out = RoPE_s( x_head(16x64) @ M(64x64) )
// One 16-row tile (= 16 heads of one (b,s) position) per wave32, 8 waves/block.
// M staged to LDS pre-swizzled in B-fragment order so each WMMA B operand is a
// single ds_load_b64 into an even-aligned VGPR pair (no fix-up movs).
// ---------------------------------------------------------------------------
#define STAGE_STRIDE 68   // rope-stage row stride (floats): conflict-free, 16B aligned
#define MSW_STRIDE  160   // swizzled-M pair-row stride (floats): 160 % 64 == 32
                          //  -> lanes 0-15 (even p) and 16-31 (odd p) use disjoint banks

__global__ __launch_bounds__(256) void rotary_wmma_kernel(
    const float* __restrict__ x,        // (B*S*16, 64) row-major
    const float* __restrict__ M,        // (64,64) from workspace
    const float* __restrict__ inv_freq, // (32,)
    float* __restrict__ out,            // same shape as x
    int ntiles)
{
    __shared__ float Msw[32 * MSW_STRIDE];        // 20480 B : swizzled M
    __shared__ float stage[8][16 * STAGE_STRIDE]; // 34816 B : per-wave rope stage

    const int tid = threadIdx.x;

    // ---- Cooperative swizzled staging of M:
    // pair-row p (=rows 2p,2p+1), col c:  Msw[p*MSW_STRIDE + 2c + {0,1}] = M[2p][c], M[2p+1][c]
    #pragma unroll
    for (int idx = tid; idx < 32 * 64; idx += 256) {
        const int p = idx >> 6;
        const int c = idx & 63;
        v2f pr;
        pr.x = M[(2 * p + 0) * H_DIM + c];
        pr.y = M[(2 * p + 1) * H_DIM + c];
        *(v2f*)(Msw + p * MSW_STRIDE + 2 * c) = pr;
    }
    __syncthreads();

    const int wave = tid >> 5;
    const int lane = tid & 31;
    const int l15  = lane & 15;
    const int hi   = lane >> 4; // 0 for lanes 0-15, 1 for lanes 16-31
    const int tile = blockIdx.x * 8 + wave;

    if (tile < ntiles) { // wave-uniform: EXEC stays all-1s inside (WMMA requirement)
        const float* __restrict__ xt = x + (size_t)tile * (16 * H_DIM);

        // ---- Load A fragments: 16 k-slices of the 16x64 input tile.
        // ISA 16x4 f32 A layout: lanes 0-15 hold K=4k,4k+1; lanes 16-31 hold K=4k+2,4k+3.
        v2f a[16];
        #pragma unroll
        for (int k = 0; k < 16; ++k)
            a[k] = *(const v2f*)(xt + l15 * H_DIM + 4 * k + 2 * hi);

        // ---- GEMM: acc[n] (16x16) = sum_k A_k (16x4) x B_{k,n} (4x16)
        // B fragment for (k,n): pair-row p = 2k + hi, column 16n + l15.
        const float* __restrict__ bbase = Msw + hi * MSW_STRIDE + 2 * l15;
        v8f acc[4];
        #pragma unroll
        for (int n = 0; n < 4; ++n) {
            v8f c = {0.f, 0.f, 0.f, 0.f, 0.f, 0.f, 0.f, 0.f};
            #pragma unroll
            for (int k = 0; k < 16; ++k) {
                const v2f b = *(const v2f*)(bbase + 2 * k * MSW_STRIDE + 32 * n);
                c = __builtin_amdgcn_wmma_f32_16x16x4_f32(
                        /*neg_a=*/false, a[k], /*neg_b=*/false, b,
                        /*c_mod=*/(short)0, c, /*reuse_a=*/false, /*reuse_b=*/false);
            }
            acc[n] = c;
        }

        // ---- Stage xr tile to per-wave LDS region.
        // C/D layout: vgpr v of acc[n] = xr[row = v + 8*hi][col = 16n + l15]
        float* __restrict__ st = stage[wave];
        #pragma unroll
        for (int n = 0; n < 4; ++n) {
            #pragma unroll
            for (int v = 0; v < 8; ++v)
                st[(v + 8 * hi) * STAGE_STRIDE + 16 * n + l15] = acc[n][v];
        }
        // Per-wave LDS ordering: DS ops are in-order within a wave; the wait +
        // memory clobber stops any reordering of the reads below.
        asm volatile("s_wait_dscnt 0" ::: "memory");

        // ---- RoPE + coalesced store.
        // Lane handles output columns [c0, c0+3] of rows (hi + 2*it), it=0..7.
        const int c0 = l15 * 4;                 // 0..60
        const bool second = (c0 >= 32);         // second half: x1*sin + x2*cos
        const int k0 = second ? (c0 - 32) : c0; // 0..28, multiple of 4
        const float pos = (float)(tile & (S_DIM - 1)); // s index

        const float an0 = pos * inv_freq[k0 + 0];
        const float an1 = pos * inv_freq[k0 + 1];
        const float an2 = pos * inv_freq[k0 + 2];
        const float an3 = pos * inv_freq[k0 + 3];
        const float sn0 = __sinf(an0), cs0 = __cosf(an0);
        const float sn1 = __sinf(an1), cs1 = __cosf(an1);
        const float sn2 = __sinf(an2), cs2 = __cosf(an2);
        const float sn3 = __sinf(an3), cs3 = __cosf(an3);

        float* __restrict__ outp = out + (size_t)tile * (16 * H_DIM);
        #pragma unroll
        for (int it = 0; it < 8; ++it) {
            const int row = hi + 2 * it;
            const float* srow = st + row * STAGE_STRIDE + 2 * k0;
            const float4 e0 = *(const float4*)(srow);
            const float4 e1 = *(const float4*)(srow + 4);
            // x1 = even cols (e*.x, e*.z), x2 = odd cols (e*.y, e*.w)
            float4 o;
            if (!second) {
                o.x = e0.x * cs0 - e0.y * sn0;
                o.y = e0.z * cs1 - e0.w * sn1;
                o.z = e1.x * cs2 - e1.y * sn2;
                o.w = e1.z * cs3 - e1.w * sn3;
            } else {
                o.x = e0.x * sn0 + e0.y * cs0;
                o.y = e0.z * sn1 + e0.w * cs1;
                o.z = e1.x * sn2 + e1.y * cs2;
                o.w = e1.z * sn3 + e1.w * cs3;
            }
            // lanes 0-15 (hi=0) write one 256B row, lanes 16-31 the next: b128 coalesced
            *(float4*)(outp + row * H_DIM + c0) = o;
        }
    }
}

// ---------------------------------------------------------------------------
extern "C" void kernel_launch(void* const* d_in, const int* in_sizes, int n_in,
                              void* d_out, int out_size, void* d_ws, size_t ws_size,
                              hipStream_t stream) {
    const float* x      = (const float*)d_in[0]; // (B,S,N_STATE)
    const float* thetas = (const float*)d_in[1]; // (32,)
    const float* pairs  = (const float*)d_in[2]; // (32,2)
    const float* tscale = (const float*)d_in[3]; // (1,)
    const float* rotm   = (const float*)d_in[4]; // (64,64)
    const float* invf   = (const float*)d_in[5]; // (32,)
    float*       out    = (float*)d_out;
    float*       M      = (float*)d_ws;          // 64*64 floats of scratch

    const int total  = in_sizes[0];              // B*S*N_STATE elements
    const int ntiles = total / (N_HEAD * H_DIM); // B*S 16x64 tiles

    build_M_kernel<<<1, 64, 0, stream>>>(thetas, pairs, tscale, rotm, M);

    const int nblocks = (ntiles + 7) / 8;        // 8 waves (tiles) per block
    rotary_wmma_kernel<<<nblocks, 256, 0, stream>>>(x, M, invf, out, ntiles);
}